// LCC_85229331022199
// MI455X (gfx1250) — compile-verified
//
#include <hip/hip_runtime.h>

// ---------------------------------------------------------------------------
// LCC loss, fused 2-pass 3x3x3 box filter for MI455X (gfx1250, wave32).
//
//  * 2D (X,Y) part of each box pass on the matrix pipe:
//      Out = BandA(16x16) * In(16x16) * BandB(16x16),  Band[i][j]=1 if 0<=i-j<=2
//    via chained V_WMMA_F32_16X16X4_F32 (exact f32; band entries are 0/1).
//  * 16^3 input tiles DMA'd into LDS by the Tensor Data Mover for fully
//    in-bounds blocks (TENSORcnt + s_wait_tensorcnt), VALU clamped loads
//    only for boundary blocks (replicate padding == index clamp).
//  * No intermediate DRAM traffic: ~113 MB total reads -> ~5 us at 23.3 TB/s.
// ---------------------------------------------------------------------------

typedef float v2f __attribute__((ext_vector_type(2)));
typedef float v8f __attribute__((ext_vector_type(8)));
typedef unsigned int u32x4 __attribute__((ext_vector_type(4)));
typedef int i32x4 __attribute__((ext_vector_type(4)));
typedef int i32x8 __attribute__((ext_vector_type(8)));

#define VOL  192
#define TILE 12            // interior output tile per block (halo 2 each side)
#define NVOX (7077888.0f)  // 192^3

__device__ __forceinline__ float band01(int d) {
  return (d >= 0 && d <= 2) ? 1.0f : 0.0f;
}

// Generic pointers to LDS carry the LDS byte address in their low 32 bits
// (ISA 10.2 aperture check).
__device__ __forceinline__ unsigned lds_addr_of(const void* p) {
  return (unsigned)(unsigned long long)p;
}

// TDM: DMA a fully in-bounds 16x16x16 f32 tile (row stride VOL, slice stride
// VOL*VOL) from global memory to contiguous LDS. D# per CDNA5 ISA ch.8.
// This toolchain exposes the 6-arg builtin (groups 0-3 + extension + cpol).
__device__ __forceinline__ void tdm_load_tile16(const float* gsrc,
                                                unsigned lds_byte) {
  unsigned long long ga = (unsigned long long)gsrc;
  u32x4 g0;
  g0[0] = 1u;                                        // count=1, user mode
  g0[1] = lds_byte;                                  // lds_addr
  g0[2] = (unsigned)ga;                              // global_addr[31:0]
  g0[3] = (unsigned)((ga >> 32) & 0x1FFFFFFu)        // global_addr[56:32]
          | (2u << 30);                              // type=2 ("image")
  i32x8 g1;
  g1[0] = (int)(2u << 16);                           // data_size=2 (4B), mask=0
  g1[1] = (int)(192u << 16);                         // tensor_dim0[15:0]=192
  g1[2] = (int)(192u << 16);                         // tensor_dim1[15:0]=192
  g1[3] = (int)(16u << 16);                          // tile_dim0=16
  g1[4] = (int)(16u | (16u << 16));                  // tile_dim1=16, tile_dim2=16
  g1[5] = 192;                                       // tensor_dim0_stride
  g1[6] = (int)(36864u << 16);                       // tensor_dim1_stride[15:0]
  g1[7] = 0;                                         // tensor_dim1_stride[47:16]
  i32x4 g2;
  g2[0] = 192;                                       // tensor_dim2
  g2[1] = 1;                                         // tensor_dim3 (unused)
  g2[2] = 192 * 192 * 192;                           // tensor_dim2_stride
  g2[3] = 0;                                         // tile_dim3=0
  i32x4 g3 = {0, 0, 0, 0};
  i32x8 g4 = {0, 0, 0, 0, 0, 0, 0, 0};               // extension slot: zeros
  __builtin_amdgcn_tensor_load_to_lds(g0, g1, g2, g3, g4, 0);
}

// 2D 3x3 box-sum of a 16x16 f32 LDS slice (row stride 16 floats), one wave.
// dst[y][x] = sum_{i=y..y+2, j=x..x+2} src[i][j]  (valid y,x in 0..13).
// WMMA f32 layouts per ISA 7.12.2:
//   A 16x4: lane L -> row M=L&15, vgpr v -> K=2*(L>>4)+v
//   B 4x16: lane L -> col N=L&15, vgpr v -> K=2*(L>>4)+v
//   C/D:    vgpr r, lane L -> (M=r+8*(L>>4), N=L&15)
__device__ __forceinline__ void box2d_wmma(const float* __restrict__ src,
                                           float* __restrict__ dst,
                                           float* __restrict__ twave) {
  const int lane = threadIdx.x & 31;
  const int r    = lane & 15;
  const int half = lane >> 4;

  v8f t = {};
#pragma unroll
  for (int kc = 0; kc < 4; ++kc) {
    const int k0 = 4 * kc + 2 * half;
    v2f a = { band01(k0 - r), band01(k0 + 1 - r) };        // BandA[r][k]
    v2f b = { src[k0 * 16 + r], src[(k0 + 1) * 16 + r] };  // In[k][r]
    t = __builtin_amdgcn_wmma_f32_16x16x4_f32(false, a, false, b,
                                              (short)0, t, false, false);
  }
  // Re-layout T via wave-private LDS (same-wave LDS ops are in-order).
#pragma unroll
  for (int vr = 0; vr < 8; ++vr) twave[(vr + 8 * half) * 16 + r] = t[vr];

  v8f o = {};
#pragma unroll
  for (int kc = 0; kc < 4; ++kc) {
    const int k0 = 4 * kc + 2 * half;
    v2f a = { twave[r * 16 + k0], twave[r * 16 + k0 + 1] };  // T[r][k]
    v2f b = { band01(k0 - r), band01(k0 + 1 - r) };          // BandB[k][r]
    o = __builtin_amdgcn_wmma_f32_16x16x4_f32(false, a, false, b,
                                              (short)0, o, false, false);
  }
#pragma unroll
  for (int vr = 0; vr < 8; ++vr) dst[(vr + 8 * half) * 16 + r] = o[vr];
}

// One block: 12^3 outputs from a 16^3 clamped tile. LDS = 120 KB:
//   [0     .. 8191 ]  sF, sM          -- phases 1-3
//   [8192  .. 16383]  b2F, b2M        -- phases 2-3
//   [16384 .. 28671]  bP1,bP2,bP3     -- phases 3-4
//   [28672 .. 30719]  wave WMMA scratch / reduction
//   [0     .. 12287]  b2P1..3 overlay -- phases 4-5
__global__ __launch_bounds__(256) void lcc_tile_kernel(
    const float* __restrict__ F, const float* __restrict__ M,
    float* __restrict__ partial) {
  __shared__ float smem[30720];
  float* sF      = smem;
  float* sM      = smem + 4096;
  float* b2F     = smem + 8192;
  float* b2M     = smem + 12288;
  float* bP      = smem + 16384;
  float* scratch = smem + 28672;
  float* b2P     = smem;  // overlay

  const int tid  = threadIdx.x;
  const int wave = tid >> 5;
  const int bx = blockIdx.x, by = blockIdx.y;
  const int bz = blockIdx.z & 15, batch = blockIdx.z >> 4;
  const long vbase = (long)batch * VOL * VOL * VOL;
  const int ox = bx * TILE - 2, oy = by * TILE - 2, oz = bz * TILE - 2;

  // -- phase 1: tile load (TDM fast path for fully in-bounds tiles) --------
  const bool interior = (ox >= 0) & (oy >= 0) & (oz >= 0) &
                        (ox + 15 < VOL) & (oy + 15 < VOL) & (oz + 15 < VOL);
  if (interior) {
    if (wave == 0) {
      const long goff = vbase + ((long)oz * VOL + oy) * VOL + ox;
      tdm_load_tile16(F + goff, lds_addr_of(sF));
      tdm_load_tile16(M + goff, lds_addr_of(sM));
      __builtin_amdgcn_s_wait_tensorcnt(0);  // wave0's TENSORcnt -> 0
    }
  } else {
    for (int i = tid; i < 4096; i += 256) {
      int x = i & 15, y = (i >> 4) & 15, z = i >> 8;
      int gx = min(max(ox + x, 0), VOL - 1);
      int gy = min(max(oy + y, 0), VOL - 1);
      int gz = min(max(oz + z, 0), VOL - 1);
      long g = vbase + ((long)gz * VOL + gy) * VOL + gx;
      sF[i] = F[g];
      sM[i] = M[g];
    }
  }
  for (int i = tid; i < 12288; i += 256) bP[i] = 0.0f;  // no NaN*0 in WMMA
  __syncthreads();  // publishes TDM/VALU-loaded tiles to all waves

  // -- phase 2: 2D box of raw F/M per slice (32 wave-tasks, WMMA) ----------
  float* tw = scratch + wave * 256;
  for (int t = wave; t < 32; t += 8) {
    int field = t & 1, d = t >> 1;
    box2d_wmma((field ? sM : sF) + d * 256, (field ? b2M : b2F) + d * 256, tw);
  }
  __syncthreads();

  // -- phase 3: local means -> residual products (14^3 interior) -----------
  const float inv27 = 1.0f / 27.0f;
  for (int i = tid; i < 14 * 14 * 14; i += 256) {
    int xx = i % 14, t2 = i / 14;
    int yy = t2 % 14, zz = t2 / 14;
    int cx = xx + 1, cy = yy + 1, cz = zz + 1;
    int bi = yy * 16 + xx;  // 2D sum at [y][x] is centered at (y+1, x+1)
    float uF = (b2F[(cz - 1) * 256 + bi] + b2F[cz * 256 + bi] +
                b2F[(cz + 1) * 256 + bi]) * inv27;
    float uM = (b2M[(cz - 1) * 256 + bi] + b2M[cz * 256 + bi] +
                b2M[(cz + 1) * 256 + bi]) * inv27;
    int si = cz * 256 + cy * 16 + cx;
    float dF = sF[si] - uF, dM = sM[si] - uM;
    bP[si]        = dF * dM;
    bP[4096 + si] = dF * dF;
    bP[8192 + si] = dM * dM;
  }
  __syncthreads();

  // -- phase 4: 2D box of products, slices z=1..14 (42 wave-tasks, WMMA) ---
  for (int t = wave; t < 42; t += 8) {
    int field = t % 3, d = 1 + t / 3;
    box2d_wmma(bP + field * 4096 + d * 256, b2P + field * 4096 + d * 256, tw);
  }
  __syncthreads();

  // -- phase 5: z = cross^2/(varF*varM+eps) on 12^3 interior, reduce -------
  float acc = 0.0f;
  for (int i = tid; i < 12 * 12 * 12; i += 256) {
    int xx = i % 12, t2 = i / 12;
    int yy = t2 % 12, zz = t2 / 12;
    int cz = zz + 2;
    int bi = (yy + 1) * 16 + (xx + 1);
    float cr = b2P[(cz - 1) * 256 + bi] + b2P[cz * 256 + bi] +
               b2P[(cz + 1) * 256 + bi];
    float vF = b2P[4096 + (cz - 1) * 256 + bi] + b2P[4096 + cz * 256 + bi] +
               b2P[4096 + (cz + 1) * 256 + bi];
    float vM = b2P[8192 + (cz - 1) * 256 + bi] + b2P[8192 + cz * 256 + bi] +
               b2P[8192 + (cz + 1) * 256 + bi];
    acc += cr * cr / (vF * vM + 1e-5f);
  }
  __syncthreads();
  scratch[tid] = acc;
  __syncthreads();
  for (int s = 128; s > 0; s >>= 1) {
    if (tid < s) scratch[tid] += scratch[tid + s];
    __syncthreads();
  }
  if (tid == 0)
    partial[(long)batch * 4096 + (bz * 256 + by * 16 + bx)] = scratch[0];
}

// Deterministic final reduction: one block per batch, fixed-order tree.
__global__ __launch_bounds__(256) void lcc_reduce_kernel(
    const float* __restrict__ partial, float* __restrict__ out) {
  __shared__ float s[256];
  const int b = blockIdx.x;
  float a = 0.0f;
  for (int i = threadIdx.x; i < 4096; i += 256) a += partial[(long)b * 4096 + i];
  s[threadIdx.x] = a;
  __syncthreads();
  for (int st = 128; st > 0; st >>= 1) {
    if (threadIdx.x < st) s[threadIdx.x] += s[threadIdx.x + st];
    __syncthreads();
  }
  if (threadIdx.x == 0) out[b] = -s[0] / NVOX;
}

extern "C" void kernel_launch(void* const* d_in, const int* in_sizes, int n_in,
                              void* d_out, int out_size, void* d_ws,
                              size_t ws_size, hipStream_t stream) {
  const float* F = (const float*)d_in[0];   // im_fixed  [2,1,192,192,192] f32
  const float* M = (const float*)d_in[1];   // im_moving [2,1,192,192,192] f32
  float* out     = (float*)d_out;           // [2] f32
  float* partial = (float*)d_ws;            // 2*4096 f32 block partials (32 KB)

  dim3 grid(16, 16, 32);  // 16^3 tiles of 12^3, x2 batches in z
  lcc_tile_kernel<<<grid, 256, 0, stream>>>(F, M, partial);
  lcc_reduce_kernel<<<2, 256, 0, stream>>>(partial, out);
}